// KoopmanLQR_3315714753078
// MI455X (gfx1250) — compile-verified
//
#include <hip/hip_runtime.h>

#define KDIM 256
#define UDIM 64

typedef float v2f __attribute__((ext_vector_type(2)));
typedef float v8f __attribute__((ext_vector_type(8)));

// ---- workspace layout (float offsets) ----
#define OFF_V    0
#define OFF_P    65536
#define OFF_ABK  131072
#define OFF_VB   196608
#define OFF_W    212992
#define OFF_M    229376
#define OFF_KG   245760      /* final K[0]  (64 x 256, row-major u,k) */
#define OFF_v    262144
#define OFF_vn   262400
#define OFF_kv   262656      /* final k[0]  (64) */
#define OFF_goal 262720
#define OFF_Qd   262976
#define OFF_Rd   263232

__device__ __forceinline__ v8f wmma4(v2f a, v2f b, v8f c) {
  // D = A(16x4,f32) * B(4x16,f32) + C(16x16,f32)
  return __builtin_amdgcn_wmma_f32_16x16x4_f32(false, a, false, b, (short)0, c,
                                               false, false);
}

__device__ __forceinline__ v8f vzero() {
  v8f z = {0.f, 0.f, 0.f, 0.f, 0.f, 0.f, 0.f, 0.f};
  return z;
}

// A-fragment: 16x4 f32. TX=true means X buffer holds X^T (Kd x M).
template <bool TX>
__device__ __forceinline__ v2f load_afrag(const float* __restrict__ X, int ldx,
                                          int mb, int k, int rc, int kk2) {
  v2f a;
  if (!TX) {
    const float* p = X + (mb + rc) * ldx + k + kk2;
    a.x = p[0]; a.y = p[1];
  } else {
    a.x = X[(k + kk2) * ldx + mb + rc];
    a.y = X[(k + kk2 + 1) * ldx + mb + rc];
  }
  return a;
}

// B-fragment: 4x16 f32 from row-major Y (Kd x N).
__device__ __forceinline__ v2f load_bfrag(const float* __restrict__ Y, int ldy,
                                          int nb, int k, int rc, int kk2) {
  v2f b;
  b.x = Y[(k + kk2) * ldy + nb + rc];
  b.y = Y[(k + kk2 + 1) * ldy + nb + rc];
  return b;
}

__device__ __forceinline__ void store_tile(float* __restrict__ D, int ldd,
                                           int mbase, int nbase, v8f acc, int lane) {
  const int col = lane & 15;
  const int r0  = (lane >> 4) * 8;
#pragma unroll
  for (int r = 0; r < 8; ++r)
    D[(mbase + r0 + r) * ldd + nbase + col] = acc[r];
}

// Single 16x16 tile (used for the small 64x64 S stage).
template <bool TX>
__device__ __forceinline__ v8f gemm_tile(const float* __restrict__ X, int ldx,
                                         const float* __restrict__ Y, int ldy,
                                         int mb, int nb, int Kd, int lane) {
  const int rc = lane & 15, kk2 = (lane >> 4) * 2;
  v8f acc = vzero();
  for (int k = 0; k < Kd; k += 4) {
    v2f a = load_afrag<TX>(X, ldx, mb, k, rc, kk2);
    v2f b = load_bfrag(Y, ldy, nb, k, rc, kk2);
    acc = wmma4(a, b, acc);
  }
  return acc;
}

// 16x32 block: 1 A-fragment shared by 2 accumulators (3 loads / 2 wmma).
template <bool TX>
__device__ __forceinline__ void gemm_block_1x2(const float* __restrict__ X, int ldx,
                                               const float* __restrict__ Y, int ldy,
                                               int mb, int nb, int Kd, int lane,
                                               v8f* acc) {
  const int rc = lane & 15, kk2 = (lane >> 4) * 2;
  for (int k = 0; k < Kd; k += 4) {
    v2f a  = load_afrag<TX>(X, ldx, mb, k, rc, kk2);
    v2f b0 = load_bfrag(Y, ldy, nb, k, rc, kk2);
    v2f b1 = load_bfrag(Y, ldy, nb + 16, k, rc, kk2);
    acc[0] = wmma4(a, b0, acc[0]);
    acc[1] = wmma4(a, b1, acc[1]);
  }
}

// 32x32 block: 2 A-frags x 2 B-frags -> 4 wmma for 4 loads (1 load/wmma).
template <bool TX>
__device__ __forceinline__ void gemm_block_2x2(const float* __restrict__ X, int ldx,
                                               const float* __restrict__ Y, int ldy,
                                               int mb, int nb, int Kd, int lane,
                                               v8f* acc) {
  const int rc = lane & 15, kk2 = (lane >> 4) * 2;
  for (int k = 0; k < Kd; k += 4) {
    v2f a0 = load_afrag<TX>(X, ldx, mb, k, rc, kk2);
    v2f a1 = load_afrag<TX>(X, ldx, mb + 16, k, rc, kk2);
    v2f b0 = load_bfrag(Y, ldy, nb, k, rc, kk2);
    v2f b1 = load_bfrag(Y, ldy, nb + 16, k, rc, kk2);
    acc[0] = wmma4(a0, b0, acc[0]);
    acc[1] = wmma4(a0, b1, acc[1]);
    acc[2] = wmma4(a1, b0, acc[2]);
    acc[3] = wmma4(a1, b1, acc[3]);
  }
}

// Async global->LDS copy (CDNA5): one b32 per lane, tracked by ASYNCcnt.
__device__ __forceinline__ void async_copy_b32(unsigned lds_byte_addr,
                                               const float* __restrict__ gbase,
                                               unsigned byte_off) {
  asm volatile("global_load_async_to_lds_b32 %0, %1, %2"
               :: "v"(lds_byte_addr), "v"(byte_off), "s"(gbase)
               : "memory");
}
__device__ __forceinline__ void wait_async0() {
  asm volatile("s_wait_asynccnt 0x0" ::: "memory");
}

// ---------------------------------------------------------------------------
// Kernel 1: backward Riccati recursion, single workgroup (32 waves on 1 WGP).
// Produces K[0] at ws+OFF_KG, k[0] at ws+OFF_kv.  All operands L2-resident.
// ---------------------------------------------------------------------------
__global__ __launch_bounds__(1024)
void riccati_kernel(const float* __restrict__ A, const float* __restrict__ B,
                    const float* __restrict__ qlog, const float* __restrict__ rlog,
                    const float* __restrict__ ggoal, const int* __restrict__ Tptr,
                    float* __restrict__ ws) {
  float* V    = ws + OFF_V;
  float* P    = ws + OFF_P;
  float* ABK  = ws + OFF_ABK;
  float* VB   = ws + OFF_VB;
  float* W    = ws + OFF_W;
  float* M    = ws + OFF_M;
  float* Kg   = ws + OFF_KG;
  float* v    = ws + OFF_v;
  float* vn   = ws + OFF_vn;
  float* kv   = ws + OFF_kv;
  float* goal = ws + OFF_goal;
  float* Qd   = ws + OFF_Qd;
  float* Rd   = ws + OFF_Rd;

  __shared__ float Saug[64][130];   // [S | I] augmented, +2 pad
  __shared__ float facs[64];
  __shared__ float pivinv;

  const int tid  = threadIdx.x;
  const int lane = tid & 31;
  const int wave = tid >> 5;
  const int T    = Tptr[0];

  // ---- init: Qd, Rd, goal_term, V = diag(Qd), v = goal_term ----
  for (int i = tid; i < KDIM * KDIM; i += 1024) V[i] = 0.0f;
  if (tid < KDIM) {
    float q = __expf(qlog[tid]);
    Qd[tid]   = q;
    goal[tid] = q * ggoal[tid];
    v[tid]    = goal[tid];
  }
  if (tid < UDIM) Rd[tid] = __expf(rlog[tid]);
  __syncthreads();
  if (tid < KDIM) V[tid * KDIM + tid] = Qd[tid];
  __syncthreads();

  for (int t = 0; t < T; ++t) {
    // ---- stage 1: VB = V @ B  (256x64, Kd=256): 32 blocks of 16x32 ----
    {
      int mb = (wave >> 1) * 16, nb = (wave & 1) * 32;
      v8f acc[2] = {vzero(), vzero()};
      gemm_block_1x2<false>(V, KDIM, B, UDIM, mb, nb, KDIM, lane, acc);
      store_tile(VB, UDIM, mb, nb, acc[0], lane);
      store_tile(VB, UDIM, mb, nb + 16, acc[1], lane);
    }
    __syncthreads();

    // ---- stage 2: S = B^T @ VB + diag(R) -> LDS, identity appended ----
    for (int tile = wave; tile < 16; tile += 32) {
      int mt = tile >> 2, nt = tile & 3;
      v8f acc = gemm_tile<true>(B, UDIM, VB, UDIM, mt * 16, nt * 16, KDIM, lane);
      int col = lane & 15, r0 = (lane >> 4) * 8;
#pragma unroll
      for (int r = 0; r < 8; ++r) {
        int rr = mt * 16 + r0 + r, cc = nt * 16 + col;
        Saug[rr][cc] = acc[r] + ((rr == cc) ? Rd[rr] : 0.0f);
      }
    }
    for (int i = tid; i < 64 * 64; i += 1024) {
      int r = i >> 6, c = i & 63;
      Saug[r][64 + c] = (r == c) ? 1.0f : 0.0f;
    }
    __syncthreads();

    // ---- stage 3: Gauss-Jordan inverse of SPD S (no pivoting needed) ----
    for (int p = 0; p < 64; ++p) {
      if (tid == 0) pivinv = 1.0f / Saug[p][p];
      if (tid < 64) facs[tid] = Saug[tid][p];
      __syncthreads();
      if (tid < 128) Saug[p][tid] *= pivinv;
      __syncthreads();
      {
        int r  = tid >> 4;
        int c0 = (tid & 15) * 8;
        if (r != p) {
          float f = facs[r];
#pragma unroll
          for (int c = 0; c < 8; ++c) Saug[r][c0 + c] -= f * Saug[p][c0 + c];
        }
      }
      __syncthreads();
    }
    // Sinv now lives in Saug[:, 64:128]

    // ---- stage 4: W = Sinv @ B^T  (64x256, Kd=64): 32 blocks of 16x32 ----
    {
      int mb = (wave >> 3) * 16, nb = (wave & 7) * 32;
      const int rc = lane & 15, kk2 = (lane >> 4) * 2;
      v8f acc0 = vzero(), acc1 = vzero();
      for (int k = 0; k < UDIM; k += 4) {
        v2f a, b0, b1;
        a.x  = Saug[mb + rc][64 + k + kk2];
        a.y  = Saug[mb + rc][64 + k + kk2 + 1];
        b0.x = B[(nb + rc) * UDIM + k + kk2];            // B^T[k][n] = B[n][k]
        b0.y = B[(nb + rc) * UDIM + k + kk2 + 1];
        b1.x = B[(nb + 16 + rc) * UDIM + k + kk2];
        b1.y = B[(nb + 16 + rc) * UDIM + k + kk2 + 1];
        acc0 = wmma4(a, b0, acc0);
        acc1 = wmma4(a, b1, acc1);
      }
      store_tile(W, KDIM, mb, nb, acc0, lane);
      store_tile(W, KDIM, mb, nb + 16, acc1, lane);
    }
    __syncthreads();

    // ---- stage 5: M = W @ V  (64x256, Kd=256): 32 blocks of 16x32 ----
    {
      int mb = (wave >> 3) * 16, nb = (wave & 7) * 32;
      v8f acc[2] = {vzero(), vzero()};
      gemm_block_1x2<false>(W, KDIM, V, KDIM, mb, nb, KDIM, lane, acc);
      store_tile(M, KDIM, mb, nb, acc[0], lane);
      store_tile(M, KDIM, mb, nb + 16, acc[1], lane);
    }
    // feed-forward kv = W @ v : only the final step's value is ever consumed
    if (t == T - 1 && tid < UDIM) {
      float s = 0.0f;
      for (int k = 0; k < KDIM; ++k) s += W[tid * KDIM + k] * v[k];
      kv[tid] = s;
    }
    __syncthreads();

    // ---- stage 6: Kg = M @ A  (64x256, Kd=256): 32 blocks of 16x32 ----
    {
      int mb = (wave >> 3) * 16, nb = (wave & 7) * 32;
      v8f acc[2] = {vzero(), vzero()};
      gemm_block_1x2<false>(M, KDIM, A, KDIM, mb, nb, KDIM, lane, acc);
      store_tile(Kg, KDIM, mb, nb, acc[0], lane);
      store_tile(Kg, KDIM, mb, nb + 16, acc[1], lane);
    }
    __syncthreads();

    // ---- stage 7: ABK = A - B @ Kg  (256x256, Kd=64): 64 blocks of 32x32 ----
    for (int blk = wave; blk < 64; blk += 32) {
      int mb = (blk >> 3) * 32, nb = (blk & 7) * 32;
      v8f acc[4] = {vzero(), vzero(), vzero(), vzero()};
      gemm_block_2x2<false>(B, UDIM, Kg, KDIM, mb, nb, UDIM, lane, acc);
      int col = lane & 15, r0 = (lane >> 4) * 8;
#pragma unroll
      for (int q = 0; q < 4; ++q) {
        int mq = mb + (q >> 1) * 16, nq = nb + (q & 1) * 16;
#pragma unroll
        for (int r = 0; r < 8; ++r) {
          int idx = (mq + r0 + r) * KDIM + nq + col;
          ABK[idx] = A[idx] - acc[q][r];
        }
      }
    }
    __syncthreads();

    // ---- stage 8: vn = ABK^T @ v + goal (tiny; thread-per-element) ----
    if (tid < KDIM) {
      float s = goal[tid];
      for (int k = 0; k < KDIM; ++k) s += ABK[k * KDIM + tid] * v[k];
      vn[tid] = s;
    }
    // ---- stage 9: P = V @ ABK  (256x256, Kd=256): 64 blocks of 32x32 ----
    for (int blk = wave; blk < 64; blk += 32) {
      int mb = (blk >> 3) * 32, nb = (blk & 7) * 32;
      v8f acc[4] = {vzero(), vzero(), vzero(), vzero()};
      gemm_block_2x2<false>(V, KDIM, ABK, KDIM, mb, nb, KDIM, lane, acc);
      store_tile(P, KDIM, mb, nb, acc[0], lane);
      store_tile(P, KDIM, mb, nb + 16, acc[1], lane);
      store_tile(P, KDIM, mb + 16, nb, acc[2], lane);
      store_tile(P, KDIM, mb + 16, nb + 16, acc[3], lane);
    }
    __syncthreads();

    // ---- stage 10: V = A^T @ P + diag(Q);  v = vn ----
    for (int blk = wave; blk < 64; blk += 32) {
      int mb = (blk >> 3) * 32, nb = (blk & 7) * 32;
      v8f acc[4] = {vzero(), vzero(), vzero(), vzero()};
      gemm_block_2x2<true>(A, KDIM, P, KDIM, mb, nb, KDIM, lane, acc);
      int col = lane & 15, r0 = (lane >> 4) * 8;
#pragma unroll
      for (int q = 0; q < 4; ++q) {
        int mq = mb + (q >> 1) * 16, nq = nb + (q & 1) * 16;
#pragma unroll
        for (int r = 0; r < 8; ++r) {
          int rr = mq + r0 + r, cc = nq + col;
          V[rr * KDIM + cc] = acc[q][r] + ((rr == cc) ? Qd[rr] : 0.0f);
        }
      }
    }
    if (tid < KDIM) v[tid] = vn[tid];
    __syncthreads();
  }
}

// ---------------------------------------------------------------------------
// Kernel 2: u = clip(k0 - g0 @ K0^T, -1, 1).  HBM-bound: g0 read exactly once.
// Block = 256 threads (8 waves) covering 128 batch rows; K0^T staged in LDS
// via CDNA5 async global->LDS copies (ASYNCcnt).
// ---------------------------------------------------------------------------
__global__ __launch_bounds__(256)
void apply_kernel(const float* __restrict__ g0, const float* __restrict__ ws,
                  float* __restrict__ out) {
  const float* K0 = ws + OFF_KG;   // [64][256] (u, k)
  const float* k0 = ws + OFF_kv;   // [64]

  __shared__ float Kt[KDIM][72];   // K0^T with pad: conflict-free B-frag reads
  __shared__ float kb[UDIM];

  const int tid  = threadIdx.x;
  const int lane = tid & 31;
  const int wave = tid >> 5;

  // Stage K0^T into LDS with async global->LDS b32 copies (per-lane scatter).
  for (int i = tid; i < UDIM * KDIM; i += 256) {
    int u = i >> 8, k = i & 255;
    unsigned ldsa = (unsigned)(size_t)&Kt[k][u];
    async_copy_b32(ldsa, K0, (unsigned)i * 4u);
  }
  if (tid < UDIM) kb[tid] = k0[tid];
  wait_async0();
  __syncthreads();

  const int rowbase = blockIdx.x * 128 + wave * 16;
  const int rc  = lane & 15;
  const int kk2 = (lane >> 4) * 2;

  const float* gp = g0 + (size_t)(rowbase + rc) * KDIM + kk2;

  v8f acc0 = vzero(), acc1 = vzero(), acc2 = vzero(), acc3 = vzero();

  for (int k = 0; k < KDIM; k += 4) {
    v2f a; a.x = gp[k]; a.y = gp[k + 1];
    __builtin_prefetch(gp + k + 64, 0, 0);
    v2f b0, b1, b2, b3;
    b0.x = Kt[k + kk2][rc];       b0.y = Kt[k + kk2 + 1][rc];
    b1.x = Kt[k + kk2][16 + rc];  b1.y = Kt[k + kk2 + 1][16 + rc];
    b2.x = Kt[k + kk2][32 + rc];  b2.y = Kt[k + kk2 + 1][32 + rc];
    b3.x = Kt[k + kk2][48 + rc];  b3.y = Kt[k + kk2 + 1][48 + rc];
    acc0 = wmma4(a, b0, acc0);
    acc1 = wmma4(a, b1, acc1);
    acc2 = wmma4(a, b2, acc2);
    acc3 = wmma4(a, b3, acc3);
  }

  const int r0 = (lane >> 4) * 8;
#pragma unroll
  for (int r = 0; r < 8; ++r) {
    float* o = out + (size_t)(rowbase + r0 + r) * UDIM;
    float u0 = kb[rc]      - acc0[r];
    float u1 = kb[16 + rc] - acc1[r];
    float u2 = kb[32 + rc] - acc2[r];
    float u3 = kb[48 + rc] - acc3[r];
    o[rc]      = fminf(1.0f, fmaxf(-1.0f, u0));
    o[16 + rc] = fminf(1.0f, fmaxf(-1.0f, u1));
    o[32 + rc] = fminf(1.0f, fmaxf(-1.0f, u2));
    o[48 + rc] = fminf(1.0f, fmaxf(-1.0f, u3));
  }
}

extern "C" void kernel_launch(void* const* d_in, const int* in_sizes, int n_in,
                              void* d_out, int out_size, void* d_ws, size_t ws_size,
                              hipStream_t stream) {
  const float* g0    = (const float*)d_in[0];
  const float* A     = (const float*)d_in[1];
  const float* B     = (const float*)d_in[2];
  const float* qlog  = (const float*)d_in[3];
  const float* rlog  = (const float*)d_in[4];
  const float* ggoal = (const float*)d_in[5];
  const int*   Tptr  = (const int*)d_in[6];

  float* ws  = (float*)d_ws;
  float* out = (float*)d_out;

  const int batch   = in_sizes[0] / KDIM;     // 131072
  const int nblocks = batch / 128;            // 1024

  riccati_kernel<<<dim3(1), dim3(1024), 0, stream>>>(A, B, qlog, rlog, ggoal, Tptr, ws);
  apply_kernel<<<dim3(nblocks), dim3(256), 0, stream>>>(g0, ws, out);
}